// GCNModel_42047729828143
// MI455X (gfx1250) — compile-verified
//
#include <hip/hip_runtime.h>

#define NUM_USERS 50000
#define NUM_ITEMS 50000
#define N_NODES   100000   // NUM_USERS + NUM_ITEMS
#define EMBED_K   64
#define N_EDGES   2000000
#define BATCH     4096

typedef __attribute__((ext_vector_type(2))) float v2f;
typedef __attribute__((ext_vector_type(8))) float v8f;

// ---------------- utility kernels ----------------

__global__ void zero_f32_kernel(float* __restrict__ p, long n) {
  long i = (long)blockIdx.x * blockDim.x + threadIdx.x;
  if (i < n) p[i] = 0.0f;
}

__global__ void relu_kernel(float* __restrict__ p, long n) {
  long i = (long)blockIdx.x * blockDim.x + threadIdx.x;
  if (i < n) p[i] = fmaxf(p[i], 0.0f);
}

// deg[col[e]] += 1
__global__ void deg_kernel(const int* __restrict__ edge_index, float* __restrict__ deg) {
  int e = blockIdx.x * blockDim.x + threadIdx.x;
  if (e < N_EDGES) {
    int c = edge_index[N_EDGES + e];   // col = edge_index[1][e]
    atomicAdd(&deg[c], 1.0f);
  }
}

// deg -> dinv in place: deg>0 ? rsqrt(deg) : 0
__global__ void dinv_kernel(float* __restrict__ deg) {
  int i = blockIdx.x * blockDim.x + threadIdx.x;
  if (i < N_NODES) {
    float d = deg[i];
    deg[i] = (d > 0.0f) ? rsqrtf(d) : 0.0f;
  }
}

// ---------------- WMMA GEMM: h = X @ W  (X is N_NODES x 64, W is 64x64) ----------------
// Block = 128 threads = 4 waves. Block b owns rows [16b, 16b+16); wave w owns cols [16w, 16w+16).
// Each wave chains 16 x V_WMMA_F32_16X16X4_F32 over K=64.
// If xb != nullptr, X is the concatenation [xa ; xb] split at NUM_USERS (tile-aligned: 50000 % 16 == 0).
__global__ void __launch_bounds__(128)
gemm_wmma_kernel(const float* __restrict__ xa, const float* __restrict__ xb,
                 const float* __restrict__ W, float* __restrict__ h) {
  const int lane = threadIdx.x & 31;
  const int wave = threadIdx.x >> 5;          // 0..3 -> column tile
  const int row0 = blockIdx.x * 16;
  const int m    = lane & 15;                 // row-in-tile for A, col-in-tile for B/C/D
  const int half = lane >> 4;                 // 0: lanes 0-15, 1: lanes 16-31

  const int node = row0 + m;
  const float* arow = (xb != nullptr && node >= NUM_USERS)
                        ? (xb + (long)(node - NUM_USERS) * EMBED_K)
                        : (xa + (long)node * EMBED_K);
  const int col = wave * 16 + m;              // output column (N index = lane&15)

  v8f acc = {};
  #pragma unroll
  for (int k0 = 0; k0 < EMBED_K; k0 += 4) {
    // A 16x4 fragment: VGPR0 = K=(k0+2*half), VGPR1 = K=(k0+2*half+1), M = lane&15
    v2f a, b;
    a.x = arow[k0 + half * 2];
    a.y = arow[k0 + half * 2 + 1];
    // B 4x16 fragment: VGPR0 = row (k0+2*half), VGPR1 = row (k0+2*half+1), N = lane&15
    b.x = W[(k0 + half * 2)     * EMBED_K + col];
    b.y = W[(k0 + half * 2 + 1) * EMBED_K + col];
    // (neg_a, A, neg_b, B, c_mod, C, reuse_a, reuse_b)
    acc = __builtin_amdgcn_wmma_f32_16x16x4_f32(false, a, false, b, (short)0, acc,
                                                false, false);
  }

  // D 16x16 f32: VGPR r -> M = r + 8*half, N = lane&15
  float* out = h + (long)(row0 + half * 8) * EMBED_K + col;
  #pragma unroll
  for (int r = 0; r < 8; ++r) out[(long)r * EMBED_K] = acc[r];
}

// ---------------- edge scatter: out[col] += h[row] * dinv[row]*dinv[col] ----------------
// One wave per edge; each lane handles features lane and lane+32 (2 f32 atomics into L2).
__global__ void scatter_kernel(const int* __restrict__ edge_index,
                               const float* __restrict__ dinv,
                               const float* __restrict__ h,
                               float* __restrict__ out) {
  long t = (long)blockIdx.x * blockDim.x + threadIdx.x;
  long e = t >> 5;
  int lane = (int)(t & 31);
  if (e >= N_EDGES) return;
  int r = edge_index[e];              // row = edge_index[0][e]
  int c = edge_index[N_EDGES + e];    // col = edge_index[1][e]
  float norm = dinv[r] * dinv[c];
  const float* hr = h + (long)r * EMBED_K;
  float* oc = out + (long)c * EMBED_K;
  atomicAdd(&oc[lane],      hr[lane]      * norm);
  atomicAdd(&oc[lane + 32], hr[lane + 32] * norm);
}

// ---------------- predictor: xui = <gu[u], gi[i]> + Bu[u] + Bi[i] + Mu ----------------
// One wave per batch element; wave32 xor-shuffle reduction.
__global__ void predict_kernel(const float* __restrict__ x,
                               const float* __restrict__ Bu,
                               const float* __restrict__ Bi,
                               const float* __restrict__ Mu,
                               const int* __restrict__ users,
                               const int* __restrict__ items,
                               float* __restrict__ out) {
  long t = (long)blockIdx.x * blockDim.x + threadIdx.x;
  long b = t >> 5;
  int lane = (int)(t & 31);
  if (b >= BATCH) return;
  int u  = users[b];
  int it = items[b];
  const float* gu = x + (long)u * EMBED_K;
  const float* gi = x + (long)(NUM_USERS + it) * EMBED_K;
  float s = gu[lane] * gi[lane] + gu[lane + 32] * gi[lane + 32];
  #pragma unroll
  for (int off = 16; off > 0; off >>= 1) s += __shfl_xor(s, off, 32);
  if (lane == 0) out[b] = s + Bu[u] + Bi[it] + Mu[0];
}

// ---------------- launch ----------------

extern "C" void kernel_launch(void* const* d_in, const int* in_sizes, int n_in,
                              void* d_out, int out_size, void* d_ws, size_t ws_size,
                              hipStream_t stream) {
  const float* Gu = (const float*)d_in[0];
  const float* Gi = (const float*)d_in[1];
  const float* Bu = (const float*)d_in[2];
  const float* Bi = (const float*)d_in[3];
  const float* Mu = (const float*)d_in[4];
  const float* W0 = (const float*)d_in[5];
  const float* W1 = (const float*)d_in[6];
  const int* edge_index = (const int*)d_in[7];
  const int* users      = (const int*)d_in[8];
  const int* items      = (const int*)d_in[9];
  float* out = (float*)d_out;

  // scratch layout (256B aligned sections)
  char* ws = (char*)d_ws;
  size_t off = 0;
  float* deg = (float*)(ws + off);                      // N_NODES floats (becomes dinv)
  off += ((size_t)N_NODES * 4 + 255) & ~(size_t)255;
  float* h   = (float*)(ws + off);                      // N_NODES x 64 floats
  off += ((size_t)N_NODES * EMBED_K * 4 + 255) & ~(size_t)255;
  float* agg = (float*)(ws + off);                      // N_NODES x 64 floats

  const long NF = (long)N_NODES * EMBED_K;

  // degrees -> dinv
  zero_f32_kernel<<<(N_NODES + 255) / 256, 256, 0, stream>>>(deg, (long)N_NODES);
  deg_kernel<<<(N_EDGES + 255) / 256, 256, 0, stream>>>(edge_index, deg);
  dinv_kernel<<<(N_NODES + 255) / 256, 256, 0, stream>>>(deg);

  // layer 1: h = [Gu;Gi] @ W0 ; agg = scatter(h) ; relu
  gemm_wmma_kernel<<<N_NODES / 16, 128, 0, stream>>>(Gu, Gi, W0, h);
  zero_f32_kernel<<<(int)((NF + 255) / 256), 256, 0, stream>>>(agg, NF);
  scatter_kernel<<<(int)(((long)N_EDGES * 32 + 255) / 256), 256, 0, stream>>>(edge_index, deg, h, agg);
  relu_kernel<<<(int)((NF + 255) / 256), 256, 0, stream>>>(agg, NF);

  // layer 2: h = agg @ W1 ; agg = scatter(h) ; relu
  gemm_wmma_kernel<<<N_NODES / 16, 128, 0, stream>>>(agg, nullptr, W1, h);
  zero_f32_kernel<<<(int)((NF + 255) / 256), 256, 0, stream>>>(agg, NF);
  scatter_kernel<<<(int)(((long)N_EDGES * 32 + 255) / 256), 256, 0, stream>>>(edge_index, deg, h, agg);
  relu_kernel<<<(int)((NF + 255) / 256), 256, 0, stream>>>(agg, NF);

  // prediction
  predict_kernel<<<(int)(((long)BATCH * 32 + 255) / 256), 256, 0, stream>>>(
      agg, Bu, Bi, Mu, users, items, out);
}